// HistogramLoss_59992103191174
// MI455X (gfx1250) — compile-verified
//
#include <hip/hip_runtime.h>
#include <stdint.h>

#define H      2048
#define HH     (H * H)          // 2^22
#define NBINS  256
#define NCH    3

typedef __attribute__((ext_vector_type(2))) float v2f;
typedef __attribute__((ext_vector_type(8))) float v8f;

__device__ __forceinline__ float de_norm255(float x) {
    float v = (x + 1.0f) * 0.5f;
    v = fminf(fmaxf(v, 0.0f), 1.0f);
    return v * 255.0f;
}

// ---------------- K1: zero workspace + output ----------------
__global__ void k_init(uint32_t* __restrict__ hd, uint32_t* __restrict__ hr,
                       uint32_t* __restrict__ touched_w, float* __restrict__ out) {
    int i = blockIdx.x * blockDim.x + threadIdx.x;
    int stride = gridDim.x * blockDim.x;
    for (int t = i; t < NCH * NBINS; t += stride) { hd[t] = 0u; hr[t] = 0u; }
    for (int t = i; t < HH / 4; t += stride) touched_w[t] = 0u;   // touched byte map
    if (i == 0) out[0] = 0.0f;
}

// ---------------- K2: gathers + LDS-privatized histograms ----------------
__global__ void k_hist(const float* __restrict__ ref, const float* __restrict__ tgt,
                       const float* __restrict__ msrc, const float* __restrict__ mtar,
                       const int* __restrict__ idx0, const int* __restrict__ idx1,
                       const int* __restrict__ idx2, const int* __restrict__ idx3,
                       uint32_t* __restrict__ g_hd, uint32_t* __restrict__ g_hr,
                       uint8_t* __restrict__ touched, int n) {
    __shared__ uint32_t sh_hd[NCH * NBINS];
    __shared__ uint32_t sh_hr[NCH * NBINS];
    int t = threadIdx.x;
    for (int j = t; j < NCH * NBINS; j += blockDim.x) { sh_hd[j] = 0u; sh_hr[j] = 0u; }
    __syncthreads();

    int stride = gridDim.x * blockDim.x;
    for (int k = blockIdx.x * blockDim.x + t; k < n; k += stride) {
        int p0 = idx0[k] * H + idx1[k];
        int p2 = idx2[k] * H + idx3[k];
        float ms = msrc[p0];
        float mt = mtar[p2];
        touched[p0] = (uint8_t)1;   // benign same-value race; duplicates write identical data
#pragma unroll
        for (int c = 0; c < NCH; ++c) {
            float dv = de_norm255(ref[c * HH + p0]) * ms;   // dst_align value
            int bd = (int)dv; if (bd > NBINS - 1) bd = NBINS - 1; if (bd < 0) bd = 0;
            atomicAdd(&sh_hd[c * NBINS + bd], 1u);          // ds_add_u32
            float rv = de_norm255(tgt[c * HH + p2]) * mt;   // ref_align value
            int br = (int)rv; if (br > NBINS - 1) br = NBINS - 1; if (br < 0) br = 0;
            atomicAdd(&sh_hr[c * NBINS + br], 1u);
        }
    }
    __syncthreads();
    for (int j = t; j < NCH * NBINS; j += blockDim.x) {
        uint32_t vd = sh_hd[j];
        uint32_t vr = sh_hr[j];
        if (vd) atomicAdd(&g_hd[j], vd);
        if (vr) atomicAdd(&g_hr[j], vr);
    }
}

// ---------------- K3: CDFs + transfer table (one block per channel) ----------------
__global__ void k_table(const uint32_t* __restrict__ g_hd, const uint32_t* __restrict__ g_hr,
                        float* __restrict__ table) {
    __shared__ float cum_d[NBINS], cum_r[NBINS];
    __shared__ float r[NBINS], a[NBINS];
    int c = blockIdx.x;
    int i = threadIdx.x;
    if (i == 0) {   // 256-element serial cumsum: negligible
        float sd = 0.0f, sr = 0.0f;
        for (int j = 0; j < NBINS; ++j) {
            sd += (float)g_hd[c * NBINS + j]; cum_d[j] = sd;
            sr += (float)g_hr[c * NBINS + j]; cum_r[j] = sr;
        }
    }
    __syncthreads();
    float totd = cum_d[NBINS - 1];
    float totr = cum_r[NBINS - 1];
    r[i] = cum_d[i] / totd;
    a[i] = cum_r[i] / totr;
    __syncthreads();

    float tbl;
    if (i == 0)               tbl = 0.0f;
    else if (i == NBINS - 1)  tbl = (float)(NBINS - 1);
    else {
        float ri = r[i];
        int found = -1;
        for (int j = 0; j < NBINS - 1; ++j) {           // j in [0,254], first hit
            if (ri >= a[j] && ri <= a[j + 1]) { found = j + 1; break; }
        }
        tbl = (found >= 0) ? (float)found : (float)i;   // not found -> arange(1,256)[i-1] == i
    }
    table[c * NBINS + i] = tbl;
}

// ---------------- K4: streaming |diff| + WMMA wave reduction ----------------
__global__ void k_loss(const float* __restrict__ inp, const float* __restrict__ ref,
                       const float* __restrict__ msrc, const uint8_t* __restrict__ touched,
                       const float* __restrict__ table, float* __restrict__ out) {
    const float scale = 1.0f / (float)(NCH * HH);
    int tid = blockIdx.x * blockDim.x + threadIdx.x;
    int stride = gridDim.x * blockDim.x;
    float s = 0.0f;
    for (int e = tid; e < NCH * HH; e += stride) {
        int c = e >> 22;            // HH == 2^22
        int p = e & (HH - 1);
        float m = msrc[p];
        float iv = de_norm255(inp[e]) * m;                 // input_masked
        float rv = de_norm255(ref[e]) * m;                 // ref_masked
        float match;
        if (touched[p]) {
            int b = (int)rv; if (b > NBINS - 1) b = NBINS - 1; if (b < 0) b = 0;
            match = table[(c << 8) + b];
        } else {
            match = rv;
        }
        s += fabsf(iv - match);
    }

    // Wave-level reduction through the matrix unit:
    // A (16x4 f32) holds the 32 per-lane partials (v2f per lane = all 64 entries),
    // B = all-ones, so D[m][n] = sum_k A[m][k] for every n. Column 0 of D,
    // i.e. colsum(lane0)+colsum(lane16), equals the sum of all A entries = wave sum.
    v2f A; A.x = s;    A.y = 0.0f;
    v2f B; B.x = 1.0f; B.y = 1.0f;
    v8f acc = {};
    acc = __builtin_amdgcn_wmma_f32_16x16x4_f32(
        /*neg_a=*/false, A, /*neg_b=*/false, B,
        /*c_mod=*/(short)0, acc, /*reuse_a=*/false, /*reuse_b=*/false);
    float colsum = acc[0] + acc[1] + acc[2] + acc[3] + acc[4] + acc[5] + acc[6] + acc[7];
    float partner = __shfl_xor(colsum, 16, 32);
    if ((threadIdx.x & 31) == 0) atomicAdd(out, (colsum + partner) * scale);
}

// ---------------- launcher ----------------
extern "C" void kernel_launch(void* const* d_in, const int* in_sizes, int n_in,
                              void* d_out, int out_size, void* d_ws, size_t ws_size,
                              hipStream_t stream) {
    const float* input_data  = (const float*)d_in[0];
    const float* target_data = (const float*)d_in[1];
    const float* ref         = (const float*)d_in[2];
    const float* mask_src    = (const float*)d_in[3];
    const float* mask_tar    = (const float*)d_in[4];
    const int*   idx0        = (const int*)d_in[5];
    const int*   idx1        = (const int*)d_in[6];
    const int*   idx2        = (const int*)d_in[7];
    const int*   idx3        = (const int*)d_in[8];
    int n = in_sizes[5];

    uint32_t* hd      = (uint32_t*)d_ws;                       // 3*256 u32
    uint32_t* hr      = hd + NCH * NBINS;                      // 3*256 u32
    float*    table   = (float*)(hr + NCH * NBINS);            // 3*256 f32
    uint8_t*  touched = (uint8_t*)(table + NCH * NBINS);       // HH bytes (4 MB)
    float*    out     = (float*)d_out;

    k_init <<<4096, 256, 0, stream>>>(hd, hr, (uint32_t*)touched, out);
    k_hist <<<1024, 256, 0, stream>>>(ref, target_data, mask_src, mask_tar,
                                      idx0, idx1, idx2, idx3, hd, hr, touched, n);
    k_table<<<NCH,  256, 0, stream>>>(hd, hr, table);
    k_loss <<<4096, 256, 0, stream>>>(input_data, ref, mask_src, touched, table, out);
}